// MixedPooling_46815143526538
// MI455X (gfx1250) — compile-verified
//
#include <hip/hip_runtime.h>
#include <hip/hip_bf16.h>
#include <math.h>
#include <stdint.h>

typedef float v4f __attribute__((ext_vector_type(4)));
typedef int   v4i __attribute__((ext_vector_type(4)));

#define WAVES 8
#define TPB   256
#define DCOLS 128
#define TROWS 16      // rows per tile (2 per wave)
#define NBUF  3       // async pipeline depth

#define AS1 __attribute__((address_space(1)))
#define AS3 __attribute__((address_space(3)))

typedef AS1 v4i* gv4i_p;   // global (addrspace 1) pointer to int4
typedef AS3 v4i* lv4i_p;   // LDS (addrspace 3) pointer to int4

#if defined(__has_builtin)
#  if __has_builtin(__builtin_amdgcn_global_load_async_to_lds_b128) && \
      __has_builtin(__builtin_amdgcn_s_wait_asynccnt)
#    define USE_ASYNC_LDS 1
#  endif
#endif
#ifndef USE_ASYNC_LDS
#  define USE_ASYNC_LDS 0
#endif

__device__ __forceinline__ v4f v4max(v4f a, v4f b) {
  v4f r;
  r.x = fmaxf(a.x, b.x);
  r.y = fmaxf(a.y, b.y);
  r.z = fmaxf(a.z, b.z);
  r.w = fmaxf(a.w, b.w);
  return r;
}

#if USE_ASYNC_LDS
__device__ __forceinline__ void async_copy_b128(const float* g, float* lds) {
  __builtin_amdgcn_global_load_async_to_lds_b128(
      (gv4i_p)(uintptr_t)g,
      (lv4i_p)(uint32_t)(uintptr_t)lds,   // low 32 bits of flat shared addr == LDS byte addr
      0, 0);
}
#endif

// Kernel 1: start[b] = lower_bound(batch, b) for b in [0, B]; start[B] == N.
__global__ void seg_starts_kernel(const int* __restrict__ batch, int N, int B,
                                  int* __restrict__ start) {
  int b = blockIdx.x * blockDim.x + threadIdx.x;
  if (b > B) return;
  int lo = 0, hi = N;
  while (lo < hi) {
    int mid = (lo + hi) >> 1;
    if (batch[mid] < b) lo = mid + 1;
    else hi = mid;
  }
  start[b] = lo;
}

// Kernel 2: one block per segment. 8 waves stream 16-row tiles through a
// 3-deep async global->LDS pipeline; deterministic serial LDS combine.
__global__ __launch_bounds__(TPB) void seg_pool_kernel(
    const float* __restrict__ x, const int* __restrict__ start,
    const float* __restrict__ alpha, float* __restrict__ out) {
  const int b = blockIdx.x;
  const int s = start[b];
  const int e = start[b + 1];
  const int wave = threadIdx.x >> 5;   // wave32
  const int lane = threadIdx.x & 31;

  v4f acc = {0.f, 0.f, 0.f, 0.f};
  v4f mx  = {-INFINITY, -INFINITY, -INFINITY, -INFINITY};

  const int nrows = e - s;

#if USE_ASYNC_LDS
  __shared__ __align__(16) float tiles[NBUF][TROWS][DCOLS];
  const int nfull = nrows >> 4;        // full 16-row tiles
  const int tail  = nrows & (TROWS - 1);
  const int rw    = wave * 2;          // this wave's rows within a tile

  // Each wave async-copies exactly the two rows it will consume -> no barriers;
  // s_wait_asynccnt (per-wave, in-order) sequences copy vs LDS read.
#define ISSUE_TILE(t, buf)                                                  \
  {                                                                         \
    const float* g0 = x + (size_t)(s + (t) * TROWS + rw) * DCOLS + lane * 4;\
    async_copy_b128(g0,         &tiles[(buf)][rw][lane * 4]);               \
    async_copy_b128(g0 + DCOLS, &tiles[(buf)][rw + 1][lane * 4]);           \
  }

  if (nfull > 0) ISSUE_TILE(0, 0);
  if (nfull > 1) ISSUE_TILE(1, 1);

  for (int t = 0; t < nfull; ++t) {
    const int buf = t % NBUF;
    if (t + 2 < nfull) {
      ISSUE_TILE(t + 2, (t + 2) % NBUF);
      __builtin_amdgcn_s_wait_asynccnt(4);   // 6 outstanding -> tile t done
    } else if (t + 1 < nfull) {
      __builtin_amdgcn_s_wait_asynccnt(2);   // 4 outstanding -> tile t done
    } else {
      __builtin_amdgcn_s_wait_asynccnt(0);   // drain
    }
    const v4f v0 = *(const v4f*)&tiles[buf][rw][lane * 4];
    const v4f v1 = *(const v4f*)&tiles[buf][rw + 1][lane * 4];
    acc += v0; mx = v4max(mx, v0);
    acc += v1; mx = v4max(mx, v1);
  }

  // Ragged tail: direct non-temporal loads, bypassing the async pipeline.
  {
    const int base = s + nfull * TROWS;
    if (rw < tail) {
      const v4f v = __builtin_nontemporal_load(
          (const v4f*)(x + (size_t)(base + rw) * DCOLS + lane * 4));
      acc += v; mx = v4max(mx, v);
    }
    if (rw + 1 < tail) {
      const v4f v = __builtin_nontemporal_load(
          (const v4f*)(x + (size_t)(base + rw + 1) * DCOLS + lane * 4));
      acc += v; mx = v4max(mx, v);
    }
  }
#undef ISSUE_TILE
#else
  for (int r = s + wave; r < e; r += WAVES) {
    const v4f v = __builtin_nontemporal_load(
        (const v4f*)(x + (size_t)r * DCOLS + lane * 4));
    acc += v;
    mx = v4max(mx, v);
  }
#endif

  // Deterministic cross-wave combine (fixed order -> bitwise stable per replay).
  __shared__ __align__(16) v4f red_s[WAVES][32];
  __shared__ __align__(16) v4f red_m[WAVES][32];
  red_s[wave][lane] = acc;
  red_m[wave][lane] = mx;
  __syncthreads();

  if (threadIdx.x < 32) {
    v4f ts = red_s[0][lane];
    v4f tm = red_m[0][lane];
#pragma unroll
    for (int w = 1; w < WAVES; ++w) {
      ts += red_s[w][lane];
      tm = v4max(tm, red_m[w][lane]);
    }
    const float cnt = (float)nrows;
    const float inv = 1.0f / fmaxf(cnt, 1.0f);
    const float a   = 1.0f / (1.0f + expf(-alpha[0]));
    const float oma = 1.0f - a;

    v4f mean = ts * inv;
    v4f m;
    m.x = __builtin_isfinite(tm.x) ? tm.x : 0.0f;
    m.y = __builtin_isfinite(tm.y) ? tm.y : 0.0f;
    m.z = __builtin_isfinite(tm.z) ? tm.z : 0.0f;
    m.w = __builtin_isfinite(tm.w) ? tm.w : 0.0f;

    v4f res;
    res.x = a * mean.x + oma * m.x;
    res.y = a * mean.y + oma * m.y;
    res.z = a * mean.z + oma * m.z;
    res.w = a * mean.w + oma * m.w;

    *(v4f*)(out + (size_t)b * DCOLS + lane * 4) = res;
  }
}

extern "C" void kernel_launch(void* const* d_in, const int* in_sizes, int n_in,
                              void* d_out, int out_size, void* d_ws, size_t ws_size,
                              hipStream_t stream) {
  (void)n_in; (void)ws_size;
  const float* x     = (const float*)d_in[0];
  const int*   batch = (const int*)d_in[1];
  const float* alpha = (const float*)d_in[2];

  const int N = in_sizes[1];
  const int D = in_sizes[0] / N;   // 128 for this problem
  const int B = out_size / D;      // 4096

  int* start = (int*)d_ws;         // (B+1) ints of scratch

  {
    const int threads = 256;
    const int blocks  = (B + 1 + threads - 1) / threads;
    hipLaunchKernelGGL(seg_starts_kernel, dim3(blocks), dim3(threads), 0, stream,
                       batch, N, B, start);
  }
  hipLaunchKernelGGL(seg_pool_kernel, dim3(B), dim3(TPB), 0, stream,
                     x, start, alpha, (float*)d_out);
}